// FeaturesEmbedding_20040317403343
// MI455X (gfx1250) — compile-verified
//
#include <hip/hip_runtime.h>
#include <stdint.h>

// ---------------------------------------------------------------------------
// Embedding gather + rating scale + sorted-segment sum for MI455X (gfx1250).
//
// Memory-bound (AI ~0.5 FLOP/B, ~111 MB moved -> ~4.8us floor @ 23.3 TB/s):
// optimize the gather path, not the math. Uses the CDNA5 async global->LDS
// engine (global_load_async_to_lds_b128 + s_wait_asynccnt) as a TMA-style
// double-buffered row gather: loads have no VGPR destinations, so each wave
// keeps a full 4KB chunk in flight while consuming the previous chunk from
// LDS. Sorted segment_ids let us keep a running per-lane partial sum and only
// touch global atomics at segment boundaries (~boundary count, not per item).
// ---------------------------------------------------------------------------

#define DIM            64
#define ROWBYTES       (DIM * 4)            // 256 B per embedding row
#define CHUNK          16                   // items staged per async chunk
#define BUFBYTES       (CHUNK * ROWBYTES)   // 4 KB per LDS buffer
#define CHUNKS_PER_WAVE 4                   // 64 items per wave -> 6400 waves
#define ITEMS_PER_WAVE (CHUNK * CHUNKS_PER_WAVE)
#define WAVES_PER_BLOCK 4
#define BLOCK_THREADS  (WAVES_PER_BLOCK * 32)

__global__ __launch_bounds__(BLOCK_THREADS)
void emb_segsum_async_kernel(const int*   __restrict__ ids,
                             const float* __restrict__ ratings,
                             const int*   __restrict__ segs,
                             const float* __restrict__ table,
                             float*       __restrict__ out,
                             int total) {
  // 4 waves * 2 buffers * 4KB = 32KB static LDS.
  __shared__ __align__(16) unsigned char stage[WAVES_PER_BLOCK * 2 * BUFBYTES];

  const int lane = threadIdx.x & 31;
  const int wv   = threadIdx.x >> 5;
  const int wave = blockIdx.x * WAVES_PER_BLOCK + wv;
  const int base_item = wave * ITEMS_PER_WAVE;
  if (base_item >= total) return;

  unsigned char* mybuf = &stage[wv * 2 * BUFBYTES];
  // Workgroup-relative LDS byte offset (low 32 bits of the generic address).
  const unsigned lds_base = (unsigned)(uintptr_t)(void*)mybuf;

  const int sub  = lane & 15;   // 16B slice within a 256B row
  const int half = lane >> 4;   // which of 2 rows this lane serves per op
  const uint64_t tbase = (uint64_t)(uintptr_t)table;

  // --- chunk metadata: lanes 0..15 hold items chunk_start..chunk_start+15 ---
  auto load_meta = [&](int chunk_start, int& mid, float& mrat, int& mseg) {
    int src = chunk_start + sub;
    int clamped = (src < total) ? src : (total - 1);
    mid  = ids[clamped];
    mseg = segs[clamped];
    mrat = (src < total) ? ratings[clamped] : 0.0f;  // OOB items contribute 0
  };

  // --- async-gather one 16-row chunk into LDS buffer `bufsel` --------------
  auto issue_async = [&](int bufsel, int mid) {
    unsigned lbase = lds_base + (unsigned)(bufsel * BUFBYTES);
#pragma unroll
    for (int p = 0; p < CHUNK / 2; ++p) {
      int item = 2 * p + half;                    // 0..15 within chunk
      int rid  = __shfl(mid, item);               // embedding row id
      uint64_t gaddr = tbase + (uint64_t)(unsigned)rid * (uint64_t)ROWBYTES
                             + (unsigned)(sub * 16);
      unsigned loff = lbase + (unsigned)(item * ROWBYTES + sub * 16);
      // CDNA5 async engine: 16B/lane -> LDS, tracked by ASYNCcnt, no VGPR dst.
      asm volatile("global_load_async_to_lds_b128 %0, %1, off"
                   :: "v"(loff), "v"(gaddr)
                   : "memory");
    }
  };

  // --- run-length accumulator: lane holds dims (2*lane, 2*lane+1) ----------
  float accx = 0.0f, accy = 0.0f;
  int cur = -1;

  auto flush = [&]() {
    if (cur >= 0) {
      atomicAdd(out + (size_t)cur * DIM + lane * 2,     accx);
      atomicAdd(out + (size_t)cur * DIM + lane * 2 + 1, accy);
    }
  };

  auto consume = [&](int bufsel, int mseg, float mrat) {
    const unsigned char* b = mybuf + bufsel * BUFBYTES;
#pragma unroll
    for (int j = 0; j < CHUNK; ++j) {
      int   bseg = __shfl(mseg, j);   // wave-uniform
      float brat = __shfl(mrat, j);
      if (bseg != cur) {              // uniform branch: segment boundary
        flush();
        cur = bseg; accx = 0.0f; accy = 0.0f;
      }
      const float2* row = (const float2*)(b + j * ROWBYTES);
      float2 v = row[lane];           // ds_load_b64, conflict-free
      accx = fmaf(v.x, brat, accx);
      accy = fmaf(v.y, brat, accy);
    }
  };

  // --- software pipeline: prefetch chunk c+1 while consuming chunk c -------
  int idA, segA; float ratA;
  int idB, segB; float ratB;
  load_meta(base_item, idA, ratA, segA);
  idB = idA; ratB = ratA; segB = segA;
  issue_async(0, idA);

  for (int c = 0; c < CHUNKS_PER_WAVE; ++c) {
    const int cursel = c & 1;
    if (c + 1 < CHUNKS_PER_WAVE) {
      load_meta(base_item + (c + 1) * CHUNK, idB, ratB, segB);
      issue_async(cursel ^ 1, idB);
      // 8 ops outstanding per chunk; async loads complete in order, so
      // ASYNCcnt <= 8 means the current chunk has fully landed in LDS.
      asm volatile("s_wait_asynccnt 0x8" ::: "memory");
    } else {
      asm volatile("s_wait_asynccnt 0x0" ::: "memory");
    }
    consume(cursel, segA, ratA);
    idA = idB; ratA = ratB; segA = segB;
  }
  flush();
}

__global__ void zero_f32_kernel(float* __restrict__ p, int n) {
  int i = blockIdx.x * blockDim.x + threadIdx.x;
  if (i < n) p[i] = 0.0f;
}

extern "C" void kernel_launch(void* const* d_in, const int* in_sizes, int n_in,
                              void* d_out, int out_size, void* d_ws, size_t ws_size,
                              hipStream_t stream) {
  (void)n_in; (void)d_ws; (void)ws_size;
  const int*   ids     = (const int*)d_in[0];
  const float* ratings = (const float*)d_in[1];
  const int*   segs    = (const int*)d_in[2];
  const float* table   = (const float*)d_in[3];
  float* out = (float*)d_out;
  const int total = in_sizes[0];

  // Atomic accumulation => output must be zeroed every launch (graph replays).
  int zt = 256;
  int zb = (out_size + zt - 1) / zt;
  hipLaunchKernelGGL(zero_f32_kernel, dim3(zb), dim3(zt), 0, stream, out, out_size);

  int nwaves  = (total + ITEMS_PER_WAVE - 1) / ITEMS_PER_WAVE;    // 6400
  int nblocks = (nwaves + WAVES_PER_BLOCK - 1) / WAVES_PER_BLOCK; // 1600
  hipLaunchKernelGGL(emb_segsum_async_kernel, dim3(nblocks), dim3(BLOCK_THREADS),
                     0, stream, ids, ratings, segs, table, out, total);
}